// DriftingModel_12661563588838
// MI455X (gfx1250) — compile-verified
//
#include <hip/hip_runtime.h>
#include <math.h>

// ---- problem dims (fixed by the reference) ----
static constexpr int Nn  = 8192;   // batch
static constexpr int Dd  = 256;    // feature dim of pos/gen
static constexpr int NDd = 128;    // z dim
static constexpr int Hh  = 1024;   // hidden
static constexpr int NT  = 16384;  // pos cols + neg cols
#define TEMPc 0.05f
#define BIGc  1000000.0f

typedef float v2f __attribute__((ext_vector_type(2)));
typedef float v8f __attribute__((ext_vector_type(8)));
typedef int   ai4 __attribute__((vector_size(4 * sizeof(int))));  // builtin's expected pointee

// ---- gfx1250 async global->LDS path (guarded; falls back to ds stores) ----
#if defined(__AMDGCN__) && __has_builtin(__builtin_amdgcn_global_load_async_to_lds_b128) && __has_builtin(__builtin_amdgcn_s_wait_asynccnt)
#define USE_ASYNC_LDS 1
#endif

__device__ __forceinline__ void async_copy16(const float* g, float* l) {
#ifdef USE_ASYNC_LDS
  __builtin_amdgcn_global_load_async_to_lds_b128(
      (__attribute__((address_space(1))) ai4*)g,
      (__attribute__((address_space(3))) ai4*)l, 0, 0);
#else
  *(float4*)l = *(const float4*)g;
#endif
}

__device__ __forceinline__ void async_join() {
#ifdef USE_ASYNC_LDS
  __builtin_amdgcn_s_wait_asynccnt(0);
#endif
}

__device__ __forceinline__ v8f wmma4(v2f a, v2f b, v8f c) {
  // V_WMMA_F32_16X16X4_F32 : D = A(16x4) x B(4x16) + C(16x16)
  return __builtin_amdgcn_wmma_f32_16x16x4_f32(false, a, false, b, (short)0, c,
                                               false, false);
}

__device__ __forceinline__ float selu_f(float v) {
  const float scale = 1.0507009873554805f;
  const float alpha = 1.6732632423543772f;
  return (v > 0.f) ? scale * v : scale * alpha * (__expf(v) - 1.f);
}

// ============================================================
// GEMM: C[N x M] = act(A[N x K] @ W[K x M] + bias), row-major.
// Block tile 64x128, 8 waves of 32x32, K-step 32, fp32 WMMA,
// double-buffered LDS fed by async global->LDS copies.
// ============================================================
__global__ __launch_bounds__(256)
void gemm_bias_act(const float* __restrict__ A, const float* __restrict__ W,
                   const float* __restrict__ bias, float* __restrict__ C,
                   int K, int M, int act) {
  __shared__ float As[2][64][36];
  __shared__ float Bs[2][32][132];
  const int tid   = threadIdx.x;
  const int lane  = tid & 31, wid = tid >> 5;
  const int waveM = wid & 1, waveN = wid >> 1;
  const int half  = lane >> 4, lr = lane & 15;
  const int tileM = blockIdx.y * 64, tileN = blockIdx.x * 128;

  auto load_tiles = [&](int buf, int k0) {
    int row = tid >> 3, c4 = (tid & 7) << 2;
    async_copy16(&A[(size_t)(tileM + row) * K + k0 + c4], &As[buf][row][c4]);
    async_copy16(&A[(size_t)(tileM + row + 32) * K + k0 + c4], &As[buf][row + 32][c4]);
#pragma unroll
    for (int it = 0; it < 4; ++it) {
      int idx = tid + it * 256;
      int r = idx >> 5, cb = (idx & 31) << 2;
      async_copy16(&W[(size_t)(k0 + r) * M + tileN + cb], &Bs[buf][r][cb]);
    }
  };

  v8f acc[2][2] = {};
  const int NK = K / 32;
  load_tiles(0, 0);

  for (int t = 0; t < NK; ++t) {
    async_join();
    __syncthreads();                 // tile t ready; buffer t-1 drained
    if (t + 1 < NK) load_tiles((t + 1) & 1, (t + 1) * 32);
    if (t + 2 < NK)                  // gfx1250 global_prefetch_b8 further ahead
      __builtin_prefetch(&A[(size_t)(tileM + (tid >> 3)) * K + (t + 2) * 32], 0, 0);
    const int buf = t & 1;
#pragma unroll
    for (int kk = 0; kk < 32; kk += 4) {
      v2f afr[2], bfr[2];
#pragma unroll
      for (int mi = 0; mi < 2; ++mi) {
        int m0 = waveM * 32 + mi * 16 + lr;
        afr[mi].x = As[buf][m0][kk + 2 * half];
        afr[mi].y = As[buf][m0][kk + 2 * half + 1];
      }
#pragma unroll
      for (int ni = 0; ni < 2; ++ni) {
        int n0 = waveN * 32 + ni * 16 + lr;
        bfr[ni].x = Bs[buf][kk + 2 * half][n0];
        bfr[ni].y = Bs[buf][kk + 2 * half + 1][n0];
      }
#pragma unroll
      for (int mi = 0; mi < 2; ++mi)
#pragma unroll
        for (int ni = 0; ni < 2; ++ni)
          acc[mi][ni] = wmma4(afr[mi], bfr[ni], acc[mi][ni]);
    }
  }

#pragma unroll
  for (int mi = 0; mi < 2; ++mi)
#pragma unroll
    for (int ni = 0; ni < 2; ++ni) {
      int col = tileN + waveN * 32 + ni * 16 + lr;
      float bv = bias[col];
#pragma unroll
      for (int r8 = 0; r8 < 8; ++r8) {
        int row = tileM + waveM * 32 + mi * 16 + (half ? 8 + r8 : r8);
        float v = acc[mi][ni][r8] + bv;
        if (act) v = selu_f(v);
        C[(size_t)row * M + col] = v;
      }
    }
}

// ============================================================
// L0[i][j] = -cdist(gen_i, Y_j)/TEMP  (Y = [pos ; gen]); diag of
// the neg half gets an extra -BIG/TEMP. A tile async, B tile is
// transposed on store (sync), same double-buffered pipeline.
// ============================================================
__global__ __launch_bounds__(256)
void logits_kernel(const float* __restrict__ gen, const float* __restrict__ pos,
                   const float* __restrict__ gn, const float* __restrict__ pn,
                   float* __restrict__ L0) {
  __shared__ float As[2][64][36];
  __shared__ float Bs[2][32][132];
  const int tid   = threadIdx.x;
  const int lane  = tid & 31, wid = tid >> 5;
  const int waveM = wid & 1, waveN = wid >> 1;
  const int half  = lane >> 4, lr = lane & 15;
  const int tileM = blockIdx.y * 64, tileN = blockIdx.x * 128;

  const float* Y = (tileN < Nn) ? pos : gen;
  const int yoff = (tileN < Nn) ? tileN : tileN - Nn;

  auto load_tiles = [&](int buf, int k0) {
    int row = tid >> 3, c4 = (tid & 7) << 2;
    async_copy16(&gen[(size_t)(tileM + row) * Dd + k0 + c4], &As[buf][row][c4]);
    async_copy16(&gen[(size_t)(tileM + row + 32) * Dd + k0 + c4], &As[buf][row + 32][c4]);
#pragma unroll
    for (int it = 0; it < 4; ++it) {  // 128 rows x 32 k, transposed into LDS
      int idx = tid + it * 256;
      int yr = idx >> 3, cc = (idx & 7) << 2;
      float4 y4 = *(const float4*)&Y[(size_t)(yoff + yr) * Dd + k0 + cc];
      Bs[buf][cc + 0][yr] = y4.x; Bs[buf][cc + 1][yr] = y4.y;
      Bs[buf][cc + 2][yr] = y4.z; Bs[buf][cc + 3][yr] = y4.w;
    }
  };

  v8f acc[2][2] = {};
  const int NK = Dd / 32;
  load_tiles(0, 0);

  for (int t = 0; t < NK; ++t) {
    async_join();
    __syncthreads();
    if (t + 1 < NK) load_tiles((t + 1) & 1, (t + 1) * 32);
    const int buf = t & 1;
#pragma unroll
    for (int kk = 0; kk < 32; kk += 4) {
      v2f afr[2], bfr[2];
#pragma unroll
      for (int mi = 0; mi < 2; ++mi) {
        int m0 = waveM * 32 + mi * 16 + lr;
        afr[mi].x = As[buf][m0][kk + 2 * half];
        afr[mi].y = As[buf][m0][kk + 2 * half + 1];
      }
#pragma unroll
      for (int ni = 0; ni < 2; ++ni) {
        int n0 = waveN * 32 + ni * 16 + lr;
        bfr[ni].x = Bs[buf][kk + 2 * half][n0];
        bfr[ni].y = Bs[buf][kk + 2 * half + 1][n0];
      }
#pragma unroll
      for (int mi = 0; mi < 2; ++mi)
#pragma unroll
        for (int ni = 0; ni < 2; ++ni)
          acc[mi][ni] = wmma4(afr[mi], bfr[ni], acc[mi][ni]);
    }
  }

#pragma unroll
  for (int mi = 0; mi < 2; ++mi)
#pragma unroll
    for (int ni = 0; ni < 2; ++ni) {
      int j = tileN + waveN * 32 + ni * 16 + lr;
      float yn = (j < Nn) ? pn[j] : gn[j - Nn];
#pragma unroll
      for (int r8 = 0; r8 < 8; ++r8) {
        int row = tileM + waveM * 32 + mi * 16 + (half ? 8 + r8 : r8);
        float d2 = gn[row] + yn - 2.f * acc[mi][ni][r8];
        float L  = -sqrtf(fmaxf(d2, 0.f)) / TEMPc;
        if (j >= Nn && (j - Nn) == row) L -= BIGc / TEMPc;
        L0[(size_t)row * NT + j] = L;
      }
    }
}

// ---- row squared-norms ----
__global__ __launch_bounds__(256)
void rownorm_kernel(const float* __restrict__ X, float* __restrict__ out, int D) {
  __shared__ float sm[256];
  int row = blockIdx.x;
  float s = 0.f;
  for (int j = threadIdx.x; j < D; j += 256) { float v = X[(size_t)row * D + j]; s += v * v; }
  sm[threadIdx.x] = s; __syncthreads();
  for (int o = 128; o > 0; o >>= 1) {
    if (threadIdx.x < o) sm[threadIdx.x] += sm[threadIdx.x + o];
    __syncthreads();
  }
  if (threadIdx.x == 0) out[row] = sm[0];
}

// ---- Sinkhorn in potential form: logit = L0 - r_i - c_j ----
__global__ __launch_bounds__(256)
void sinkhorn_row(const float* __restrict__ L0, const float* __restrict__ c,
                  float* __restrict__ r) {
  __shared__ float sm[256], ss[256];
  int i = blockIdx.x;
  float m = -INFINITY, s = 0.f;
  for (int j = threadIdx.x; j < NT; j += 256) {
    float x = L0[(size_t)i * NT + j] - c[j];
    if (x > m) { s = s * __expf(m - x) + 1.f; m = x; } else s += __expf(x - m);
  }
  sm[threadIdx.x] = m; ss[threadIdx.x] = s; __syncthreads();
  for (int o = 128; o > 0; o >>= 1) {
    if (threadIdx.x < o) {
      float m2 = sm[threadIdx.x + o], s2 = ss[threadIdx.x + o];
      float M = fmaxf(sm[threadIdx.x], m2);
      ss[threadIdx.x] = ss[threadIdx.x] * __expf(sm[threadIdx.x] - M) + s2 * __expf(m2 - M);
      sm[threadIdx.x] = M;
    }
    __syncthreads();
  }
  if (threadIdx.x == 0) r[i] = sm[0] + __logf(ss[0]);
}

__global__ __launch_bounds__(256)
void sinkhorn_col(const float* __restrict__ L0, const float* __restrict__ r,
                  float* __restrict__ c) {
  int j = blockIdx.x * 256 + threadIdx.x;
  float m = -INFINITY, s = 0.f;
  for (int i = 0; i < Nn; ++i) {
    float x = L0[(size_t)i * NT + j] - r[i];
    if (x > m) { s = s * __expf(m - x) + 1.f; m = x; } else s += __expf(x - m);
  }
  c[j] = m + __logf(s);
}

// ---- per-row sums of A over pos / neg halves ----
__global__ __launch_bounds__(256)
void rowsums_kernel(const float* __restrict__ L0, const float* __restrict__ r,
                    const float* __restrict__ c, float* __restrict__ sp,
                    float* __restrict__ sn) {
  __shared__ float smp[256], smn[256];
  int i = blockIdx.x;
  float rr = r[i], ap = 0.f, an = 0.f;
  for (int j = threadIdx.x; j < NT; j += 256) {
    float a = __expf(L0[(size_t)i * NT + j] - rr - c[j]);
    if (j < Nn) ap += a; else an += a;
  }
  smp[threadIdx.x] = ap; smn[threadIdx.x] = an; __syncthreads();
  for (int o = 128; o > 0; o >>= 1) {
    if (threadIdx.x < o) {
      smp[threadIdx.x] += smp[threadIdx.x + o];
      smn[threadIdx.x] += smn[threadIdx.x + o];
    }
    __syncthreads();
  }
  if (threadIdx.x == 0) { sp[i] = smp[0]; sn[i] = smn[0]; }
}

__global__ void zero_kernel(float* __restrict__ p, int n) {
  int i = blockIdx.x * 256 + threadIdx.x;
  if (i < n) p[i] = 0.f;
}

// ============================================================
// Drift GEMM:  V = Wpos@pos - Wneg@gen with the A operand
// reconstructed on the fly: a = exp(L0 - r - c) * (sn | -sp).
// A tile: VALU transform + ds stores. B (Y) tile: async copies.
// Output: out_i += sum_d V_id^2 (cross-block atomicAdd).
// ============================================================
__global__ __launch_bounds__(256)
void drift_kernel(const float* __restrict__ L0, const float* __restrict__ r,
                  const float* __restrict__ c, const float* __restrict__ sp,
                  const float* __restrict__ sn, const float* __restrict__ pos,
                  const float* __restrict__ gen, float* __restrict__ out) {
  __shared__ float As[2][64][36];
  __shared__ float Bs[2][32][132];
  const int tid   = threadIdx.x;
  const int lane  = tid & 31, wid = tid >> 5;
  const int waveM = wid & 1, waveN = wid >> 1;
  const int half  = lane >> 4, lr = lane & 15;
  const int tileM = blockIdx.y * 64, tileN = blockIdx.x * 128;

  auto load_tiles = [&](int buf, int k0) {
    { // A tile: 64 rows x 32 j-cols, exp-transform on load
      int rw = tid >> 3, c4 = (tid & 7) << 2;
#pragma unroll
      for (int h = 0; h < 2; ++h) {
        int row  = rw + h * 32;
        int grow = tileM + row;
        float rr = r[grow];
        float sc = (k0 < Nn) ? sn[grow] : -sp[grow];
        float4 l4 = *(const float4*)&L0[(size_t)grow * NT + k0 + c4];
        float4 cv = *(const float4*)&c[k0 + c4];
        As[buf][row][c4 + 0] = __expf(l4.x - rr - cv.x) * sc;
        As[buf][row][c4 + 1] = __expf(l4.y - rr - cv.y) * sc;
        As[buf][row][c4 + 2] = __expf(l4.z - rr - cv.z) * sc;
        As[buf][row][c4 + 3] = __expf(l4.w - rr - cv.w) * sc;
      }
    }
    { // B tile: Y rows k0..k0+31, feature cols tileN..tileN+127
      const float* Y = (k0 < Nn) ? pos : gen;
      int yoff = (k0 < Nn) ? k0 : k0 - Nn;
#pragma unroll
      for (int it = 0; it < 4; ++it) {
        int idx = tid + it * 256;
        int rw = idx >> 5, cc = (idx & 31) << 2;
        async_copy16(&Y[(size_t)(yoff + rw) * Dd + tileN + cc], &Bs[buf][rw][cc]);
      }
    }
  };

  v8f acc[2][2] = {};
  const int NK = NT / 32;
  load_tiles(0, 0);

  for (int t = 0; t < NK; ++t) {
    async_join();
    __syncthreads();
    if (t + 1 < NK) load_tiles((t + 1) & 1, (t + 1) * 32);
    if (t + 2 < NK)
      __builtin_prefetch(&L0[(size_t)(tileM + (tid >> 3)) * NT + (t + 2) * 32], 0, 0);
    const int buf = t & 1;
#pragma unroll
    for (int kk = 0; kk < 32; kk += 4) {
      v2f afr[2], bfr[2];
#pragma unroll
      for (int mi = 0; mi < 2; ++mi) {
        int m0 = waveM * 32 + mi * 16 + lr;
        afr[mi].x = As[buf][m0][kk + 2 * half];
        afr[mi].y = As[buf][m0][kk + 2 * half + 1];
      }
#pragma unroll
      for (int ni = 0; ni < 2; ++ni) {
        int n0 = waveN * 32 + ni * 16 + lr;
        bfr[ni].x = Bs[buf][kk + 2 * half][n0];
        bfr[ni].y = Bs[buf][kk + 2 * half + 1][n0];
      }
#pragma unroll
      for (int mi = 0; mi < 2; ++mi)
#pragma unroll
        for (int ni = 0; ni < 2; ++ni)
          acc[mi][ni] = wmma4(afr[mi], bfr[ni], acc[mi][ni]);
    }
  }

  // out_i = sum_d V_id^2 ; reduce 16 lanes sharing a row, atomic across blocks
#pragma unroll
  for (int mi = 0; mi < 2; ++mi) {
#pragma unroll
    for (int r8 = 0; r8 < 8; ++r8) {
      int row = tileM + waveM * 32 + mi * 16 + (half ? 8 + r8 : r8);
      float v = acc[mi][0][r8] * acc[mi][0][r8] + acc[mi][1][r8] * acc[mi][1][r8];
#pragma unroll
      for (int o = 1; o < 16; o <<= 1) v += __shfl_xor(v, o, 32);
      if (lr == 0) atomicAdd(&out[row], v);
    }
  }
}

// ============================================================
extern "C" void kernel_launch(void* const* d_in, const int* in_sizes, int n_in,
                              void* d_out, int out_size, void* d_ws, size_t ws_size,
                              hipStream_t stream) {
  const float* pos = (const float*)d_in[0];
  const float* z   = (const float*)d_in[1];
  const float* W1  = (const float*)d_in[2];
  const float* b1  = (const float*)d_in[3];
  const float* W2  = (const float*)d_in[4];
  const float* b2  = (const float*)d_in[5];
  const float* W3  = (const float*)d_in[6];
  const float* b3  = (const float*)d_in[7];
  const float* W4  = (const float*)d_in[8];
  const float* b4  = (const float*)d_in[9];
  const float* W5  = (const float*)d_in[10];
  const float* b5  = (const float*)d_in[11];
  float* out = (float*)d_out;

  float* ws = (float*)d_ws;
  size_t off = 0;
  float* h1  = ws + off; off += (size_t)Nn * Hh;
  float* h2  = ws + off; off += (size_t)Nn * Hh;
  float* gen = ws + off; off += (size_t)Nn * Dd;
  float* L0  = ws + off; off += (size_t)Nn * NT;
  float* gn  = ws + off; off += Nn;
  float* pn  = ws + off; off += Nn;
  float* rr  = ws + off; off += Nn;
  float* cc  = ws + off; off += NT;
  float* sp  = ws + off; off += Nn;
  float* sn  = ws + off; off += Nn;

  dim3 blk(256);

  // MLP (SELU on layers 1..4)
  gemm_bias_act<<<dim3(Hh / 128, Nn / 64), blk, 0, stream>>>(z,  W1, b1, h1, NDd, Hh, 1);
  gemm_bias_act<<<dim3(Hh / 128, Nn / 64), blk, 0, stream>>>(h1, W2, b2, h2, Hh,  Hh, 1);
  gemm_bias_act<<<dim3(Hh / 128, Nn / 64), blk, 0, stream>>>(h2, W3, b3, h1, Hh,  Hh, 1);
  gemm_bias_act<<<dim3(Hh / 128, Nn / 64), blk, 0, stream>>>(h1, W4, b4, h2, Hh,  Hh, 1);
  gemm_bias_act<<<dim3(Dd / 128, Nn / 64), blk, 0, stream>>>(h2, W5, b5, gen, Hh, Dd, 0);

  // distances -> L0
  rownorm_kernel<<<Nn, blk, 0, stream>>>(gen, gn, Dd);
  rownorm_kernel<<<Nn, blk, 0, stream>>>(pos, pn, Dd);
  logits_kernel<<<dim3(NT / 128, Nn / 64), blk, 0, stream>>>(gen, pos, gn, pn, L0);

  // Sinkhorn (potential form; L0 stays immutable)
  zero_kernel<<<NT / 256, blk, 0, stream>>>(cc, NT);
  for (int it = 0; it < 5; ++it) {
    sinkhorn_row<<<Nn, blk, 0, stream>>>(L0, cc, rr);
    sinkhorn_col<<<NT / 256, blk, 0, stream>>>(L0, rr, cc);
  }
  rowsums_kernel<<<Nn, blk, 0, stream>>>(L0, rr, cc, sp, sn);

  // drift GEMM + squared-norm output
  zero_kernel<<<Nn / 256, blk, 0, stream>>>(out, Nn);
  drift_kernel<<<dim3(Dd / 128, Nn / 64), blk, 0, stream>>>(L0, rr, cc, sp, sn, pos, gen, out);
}